// KNRM_63891933495509
// MI455X (gfx1250) — compile-verified
//
#include <hip/hip_runtime.h>
#include <math.h>
#include <stdint.h>

// ---------------------------------------------------------------------------
// KNRM scorer for MI455X (gfx1250, wave32).
//  - fp32 WMMA 16x16x4 for the 32x256 (K=128) cosine GEMM (reference-matching
//    numerics; workload is L2-gather bound, so fp32 matrix path is free).
//  - async global->LDS staging (GLOBAL_LOAD_ASYNC_TO_LDS_B128, ASYNCcnt).
//  - register-resident Gaussian-kernel accumulators across doc chunks;
//    one cross-lane reduce per pair via v_permlane16 (VALU, not LDS).
// ---------------------------------------------------------------------------

typedef __attribute__((ext_vector_type(2))) float v2f;
typedef __attribute__((ext_vector_type(8))) float v8f;
typedef int v4i __attribute__((vector_size(4 * sizeof(int))));

#define QLEN    32
#define DLEN    256
#define EMBED   128
#define KNUM    11
#define CHUNK   64
#define LSTRIDE 132   // 128 + 4 pad: breaks the 64-bank row alias

__constant__ float c_mus[KNUM] = {-0.9f,-0.7f,-0.5f,-0.3f,-0.1f,0.1f,0.3f,0.5f,0.7f,0.9f,1.0f};
// C_k = -0.5 * (1/sigma)^2 * log2(e):  exp(-0.5 z^2) == exp2(C_k * (s-mu)^2)
__constant__ float c_e2c[KNUM] = {
    -72.13475204f,-72.13475204f,-72.13475204f,-72.13475204f,-72.13475204f,
    -72.13475204f,-72.13475204f,-72.13475204f,-72.13475204f,-72.13475204f,
    -721347.5f};

#if defined(__gfx1250__) && __has_builtin(__builtin_amdgcn_global_load_async_to_lds_b128)
#define HAVE_ASYNC_LDS 1
#else
#define HAVE_ASYNC_LDS 0
#endif

// copy 16 bytes global -> LDS (async path when available)
__device__ __forceinline__ void stage16(const float* __restrict__ gsrc, float* ldst) {
#if HAVE_ASYNC_LDS
    __builtin_amdgcn_global_load_async_to_lds_b128(
        (__attribute__((address_space(1))) v4i*)(uintptr_t)gsrc,
        (__attribute__((address_space(3))) v4i*)(unsigned)(uintptr_t)ldst,
        0, 0);
#else
    *reinterpret_cast<float4*>(ldst) = *reinterpret_cast<const float4*>(gsrc);
#endif
}

__device__ __forceinline__ void stage_wait() {
#if HAVE_ASYNC_LDS
#if __has_builtin(__builtin_amdgcn_s_wait_asynccnt)
    __builtin_amdgcn_s_wait_asynccnt(0);
#else
    asm volatile("s_wait_asynccnt 0x0" ::: "memory");
#endif
#endif
}

__device__ __forceinline__ float fast_rcp(float x) {
#if __has_builtin(__builtin_amdgcn_rcpf)
    return __builtin_amdgcn_rcpf(x);
#else
    return 1.0f / x;
#endif
}

__device__ __forceinline__ float fast_exp2(float x) {
#if __has_builtin(__builtin_amdgcn_exp2f)
    return __builtin_amdgcn_exp2f(x);
#else
    return exp2f(x);
#endif
}

// sum across the 16 lanes of each half-wave (xor 1,2,4,8 inside 16-lane rows)
__device__ __forceinline__ float rsum16(float x) {
#if __has_builtin(__builtin_amdgcn_permlane16)
    union { float f; unsigned u; } a, r;
    // xor 1
    a.f = x; r.u = __builtin_amdgcn_permlane16(a.u, a.u, 0x67452301u, 0xEFCDAB89u, false, false); x += r.f;
    // xor 2
    a.f = x; r.u = __builtin_amdgcn_permlane16(a.u, a.u, 0x54761032u, 0xDCFE98BAu, false, false); x += r.f;
    // xor 4
    a.f = x; r.u = __builtin_amdgcn_permlane16(a.u, a.u, 0x32107654u, 0xBA98FEDCu, false, false); x += r.f;
    // xor 8
    a.f = x; r.u = __builtin_amdgcn_permlane16(a.u, a.u, 0xFEDCBA98u, 0x76543210u, false, false); x += r.f;
    return x;
#else
    x += __shfl_xor(x, 1);
    x += __shfl_xor(x, 2);
    x += __shfl_xor(x, 4);
    x += __shfl_xor(x, 8);
    return x;
#endif
}

__global__ __launch_bounds__(256) void knrm_kernel(
    const float* __restrict__ emb,
    const float* __restrict__ mlp_w,
    const float* __restrict__ mlp_b,
    const long long* __restrict__ q1,
    const long long* __restrict__ d1,
    const long long* __restrict__ q2,
    const long long* __restrict__ d2,
    float* __restrict__ out)
{
    __shared__ float sQ[QLEN * LSTRIDE];    // 16.9 KB query rows
    __shared__ float sD[CHUNK * LSTRIDE];   // 33.8 KB doc-chunk rows
    __shared__ float sqn[QLEN];
    __shared__ float sdn[CHUNK];
    __shared__ float sS[QLEN * KNUM];       // sum over doc of kernel values
    __shared__ float sLogit[2];

    const int tid  = threadIdx.x;
    const int b    = blockIdx.x;
    const int lane = tid & 31;
    const int wave = tid >> 5;
    const int half = lane >> 4;   // K-pair selector in f32 WMMA layout
    const int lp   = lane & 15;   // M (A) / N (B) row within tile
    const int m_t  = wave & 1;    // 2 row tiles (Q=32)
    const int n_t  = wave >> 1;   // 4 col tiles (chunk=64)

    for (int pair = 0; pair < 2; ++pair) {
        const long long* qtok = (pair ? q2 : q1) + (size_t)b * QLEN;
        const long long* dtok = (pair ? d2 : d1) + (size_t)b * DLEN;

        // ---- stage query embeddings (async gather, L2-resident table) ----
        for (int f = tid; f < QLEN * (EMBED / 4); f += 256) {
            int row = f >> 5, c4 = f & 31;
            size_t tok = (size_t)qtok[row];
            stage16(emb + tok * EMBED + c4 * 4, &sQ[row * LSTRIDE + c4 * 4]);
        }
        for (int f = tid; f < QLEN * KNUM; f += 256) sS[f] = 0.f;
        stage_wait();
        __syncthreads();

        if (tid < QLEN) {
            float s = 0.f;
            const float4* r = reinterpret_cast<const float4*>(&sQ[tid * LSTRIDE]);
            #pragma unroll 8
            for (int e = 0; e < EMBED / 4; ++e) {
                float4 v = r[e];
                s += v.x * v.x + v.y * v.y + v.z * v.z + v.w * v.w;
            }
            sqn[tid] = sqrtf(s);
        }
        __syncthreads();

        // this lane's 8 query-row norms are invariant across doc chunks
        float qn_r[8];
        #pragma unroll
        for (int v = 0; v < 8; ++v) qn_r[v] = sqn[m_t * 16 + v + 8 * half];

        // register-resident kernel-sum accumulators (sum over this lane's
        // doc column across all chunks), reduced across lanes once at the end
        float acc[8][KNUM];
        #pragma unroll
        for (int v = 0; v < 8; ++v)
            #pragma unroll
            for (int kk = 0; kk < KNUM; ++kk) acc[v][kk] = 0.f;

        for (int chunk = 0; chunk < DLEN / CHUNK; ++chunk) {
            // ---- stage doc chunk (async gather) ----
            for (int f = tid; f < CHUNK * (EMBED / 4); f += 256) {
                int row = f >> 5, c4 = f & 31;
                size_t tok = (size_t)dtok[chunk * CHUNK + row];
                stage16(emb + tok * EMBED + c4 * 4, &sD[row * LSTRIDE + c4 * 4]);
            }
            stage_wait();
            __syncthreads();
            if (tid < CHUNK) {
                float s = 0.f;
                const float4* r = reinterpret_cast<const float4*>(&sD[tid * LSTRIDE]);
                #pragma unroll 8
                for (int e = 0; e < EMBED / 4; ++e) {
                    float4 v = r[e];
                    s += v.x * v.x + v.y * v.y + v.z * v.z + v.w * v.w;
                }
                sdn[tid] = sqrtf(s);
            }
            __syncthreads();

            // ---- fp32 WMMA: C[16x16] += A[16x4] * B[4x16], K = 128 ----
            // A frag: lane lp holds row M=lp; VGPR v = K = 2*half + v
            // B frag: lane lp holds col N=lp; VGPR v = K = 2*half + v
            v8f c = {};
            const float* aBase = &sQ[(m_t * 16 + lp) * LSTRIDE + half * 2];
            const float* bBase = &sD[(n_t * 16 + lp) * LSTRIDE + half * 2];
            #pragma unroll 8
            for (int k = 0; k < EMBED; k += 4) {
                float2 av = *reinterpret_cast<const float2*>(aBase + k);
                float2 bv = *reinterpret_cast<const float2*>(bBase + k);
                v2f a, bb;
                a.x = av.x;  a.y = av.y;
                bb.x = bv.x; bb.y = bv.y;
                c = __builtin_amdgcn_wmma_f32_16x16x4_f32(
                        false, a, false, bb, (short)0, c, false, false);
            }

            // ---- cosine -> 11 Gaussian kernels, accumulate in registers ----
            // C/D layout: VGPR v = row M = v + 8*half (lanes hold N = lp)
            const float dnv = sdn[n_t * 16 + lp];
            #pragma unroll
            for (int v = 0; v < 8; ++v) {
                float inv = fast_rcp(fmaxf(qn_r[v] * dnv, 1e-8f));
                float sim = c[v] * inv;
                #pragma unroll
                for (int kk = 0; kk < KNUM; ++kk) {
                    float d = sim - c_mus[kk];
                    acc[v][kk] += fast_exp2((c_e2c[kk] * d) * d);
                }
            }
            __syncthreads();
        }

        // ---- one cross-lane reduce per pair, then scatter into sS ----
        #pragma unroll
        for (int v = 0; v < 8; ++v) {
            int m = m_t * 16 + v + 8 * half;
            #pragma unroll
            for (int kk = 0; kk < KNUM; ++kk) {
                float val = rsum16(acc[v][kk]);
                if (lp == 0) atomicAdd(&sS[m * KNUM + kk], val);
            }
        }
        __syncthreads();

        // ---- pooled = sum_q log1p(S); logit = pooled . w + b ----
        if (wave == 0) {
            float a = 0.f;
            if (lane < KNUM) {
                float w = mlp_w[lane];
                float p = 0.f;
                for (int q = 0; q < QLEN; ++q) p += log1pf(sS[q * KNUM + lane]);
                a = p * w;
            }
            a = rsum16(a);
            if (lane == 0) sLogit[pair] = a + mlp_b[0];
        }
        __syncthreads();
    }

    if (tid == 0) {
        float x = sLogit[0] - sLogit[1];
        out[b] = 1.f / (1.f + __expf(-x));
    }
}

extern "C" void kernel_launch(void* const* d_in, const int* in_sizes, int n_in,
                              void* d_out, int out_size, void* d_ws, size_t ws_size,
                              hipStream_t stream) {
    (void)n_in; (void)out_size; (void)d_ws; (void)ws_size;
    const float*     emb   = (const float*)d_in[0];
    const float*     mlp_w = (const float*)d_in[1];
    const float*     mlp_b = (const float*)d_in[2];
    const long long* q1    = (const long long*)d_in[3];
    const long long* d1    = (const long long*)d_in[4];
    const long long* q2    = (const long long*)d_in[5];
    const long long* d2    = (const long long*)d_in[6];
    float* out = (float*)d_out;

    const int B = in_sizes[3] / QLEN;   // 1024
    knrm_kernel<<<B, 256, 0, stream>>>(emb, mlp_w, mlp_b, q1, d1, q2, d2, out);
}